// RotationalQuantizer_21328807592116
// MI455X (gfx1250) — compile-verified
//
#include <hip/hip_runtime.h>

// Rotational quantizer for MI455X (gfx1250, wave32).
// B=4096 rows, C=256 codes, D=128 dims, all fp32.
//
// Heavy op: S = XC * Codes^T  (4096x128 @ 128x256) done with
// V_WMMA_F32_16X16X4_F32 (full fp32 tensor path, matches reference precision).
// Rotation R is rank-2 perturbed identity -> applied analytically per row.

#define B_ROWS 4096
#define N_CODES 256
#define DIM 128
#define INV_SQRT_D 0.08838834764831845f   // 1/sqrt(128)
#define EPS_F 1e-6f

typedef __attribute__((ext_vector_type(2))) float v2f;
typedef __attribute__((ext_vector_type(8))) float v8f;

__device__ __forceinline__ float waveSum(float v) {
    v += __shfl_xor(v, 1, 32);
    v += __shfl_xor(v, 2, 32);
    v += __shfl_xor(v, 4, 32);
    v += __shfl_xor(v, 8, 32);
    v += __shfl_xor(v, 16, 32);
    return v;
}

// Kernel 1: one wave per row. xc = R^T x; also emit u (unit prev_q) and c = u.v
__global__ void rot_canonical_kernel(const float* __restrict__ x,
                                     const float* __restrict__ pq,
                                     float* __restrict__ xc,
                                     float* __restrict__ uOut,
                                     float* __restrict__ cOut) {
    const int wave = threadIdx.x >> 5;
    const int lane = threadIdx.x & 31;
    const int b = blockIdx.x * (blockDim.x >> 5) + wave;
    if (b >= B_ROWS) return;

    const float4 xi = reinterpret_cast<const float4*>(x  + b * DIM)[lane];
    const float4 pi = reinterpret_cast<const float4*>(pq + b * DIM)[lane];

    float pn2 = waveSum(pi.x*pi.x + pi.y*pi.y + pi.z*pi.z + pi.w*pi.w);
    float inv = 1.0f / fmaxf(sqrtf(pn2), EPS_F);
    float4 u;
    u.x = pi.x * inv; u.y = pi.y * inv; u.z = pi.z * inv; u.w = pi.w * inv;

    float c   = waveSum(u.x + u.y + u.z + u.w) * INV_SQRT_D;        // u.v
    float s_u = waveSum(u.x*xi.x + u.y*xi.y + u.z*xi.z + u.w*xi.w); // u.x
    float s_v = waveSum(xi.x + xi.y + xi.z + xi.w) * INV_SQRT_D;    // v.x
    float t = 1.0f + c + EPS_F;
    // R^T x = x + u*(-s_v + (c*s_v - s_u)/t) + v*(s_u + (c*s_u - s_v)/t)
    float cu = -s_v + (c * s_v - s_u) / t;
    float cv = ( s_u + (c * s_u - s_v) / t) * INV_SQRT_D;           // fold v_i

    float4 o;
    o.x = xi.x + u.x * cu + cv;
    o.y = xi.y + u.y * cu + cv;
    o.z = xi.z + u.z * cu + cv;
    o.w = xi.w + u.w * cu + cv;
    reinterpret_cast<float4*>(xc   + b * DIM)[lane] = o;
    reinterpret_cast<float4*>(uOut + b * DIM)[lane] = u;
    if (lane == 0) cOut[b] = c;
}

// Kernel 2: one wave per code: ||code_n||^2
__global__ void code_norm_kernel(const float* __restrict__ codes,
                                 float* __restrict__ cn) {
    const int n = blockIdx.x;
    const int lane = threadIdx.x;
    float4 v = reinterpret_cast<const float4*>(codes + n * DIM)[lane];
    float s = waveSum(v.x*v.x + v.y*v.y + v.z*v.z + v.w*v.w);
    if (lane == 0) cn[n] = s;
}

// Kernel 3: per block: 16-row XC tile vs all 256 codes via fp32 WMMA, argmin.
// 512 threads = 16 waves; wave w owns the 16-code tile n0 = 16*w.
__global__ void dist_argmin_kernel(const float* __restrict__ xc,
                                   const float* __restrict__ codes,
                                   const float* __restrict__ cn,
                                   int*   __restrict__ idxOut,
                                   float* __restrict__ idxOutF) {
    __shared__ float As[16 * 132];       // 16 rows x 128, padded stride 132
    __shared__ float cnS[N_CODES];
    __shared__ float minV[16][16];       // [m][wave]
    __shared__ int   minI[16][16];

    const int tid  = threadIdx.x;        // 0..511
    const int wave = tid >> 5;           // 0..15
    const int lane = tid & 31;
    const int m0   = blockIdx.x * 16;

    {   // stage A tile: each thread moves one float4
        const int row = tid >> 5;
        const int col = (tid & 31) * 4;
        float4 v = *reinterpret_cast<const float4*>(xc + (m0 + row) * DIM + col);
        As[row * 132 + col + 0] = v.x;
        As[row * 132 + col + 1] = v.y;
        As[row * 132 + col + 2] = v.z;
        As[row * 132 + col + 3] = v.w;
    }
    if (tid < N_CODES) cnS[tid] = cn[tid];
    __syncthreads();

    const int half = lane >> 4;          // 0: K=k0..k0+1, 1: K=k0+2..k0+3
    const int l16  = lane & 15;
    const int n0   = wave * 16;
    const float* __restrict__ codeRow = codes + (n0 + l16) * DIM;
    const float* __restrict__ aRow    = As + l16 * 132;

    v8f acc = {};
    for (int k0 = 0; k0 < DIM; k0 += 4) {
        const int ka = k0 + half * 2;
        v2f a, bfr;
        a[0]   = aRow[ka];     a[1]   = aRow[ka + 1];
        bfr[0] = codeRow[ka];  bfr[1] = codeRow[ka + 1];
        acc = __builtin_amdgcn_wmma_f32_16x16x4_f32(
                  false, a, false, bfr, (short)0, acc, false, false);
    }

    // score(m,n) = ||code_n||^2 - 2*dot ; argmin over n (ties -> lowest index)
    const float cterm = cnS[n0 + l16];
#pragma unroll
    for (int r = 0; r < 8; ++r) {
        const int m = r + half * 8;
        float score = cterm - 2.0f * acc[r];
        int   idx   = n0 + l16;
#pragma unroll
        for (int msk = 1; msk <= 8; msk <<= 1) {
            float s2 = __shfl_xor(score, msk, 32);
            int   i2 = __shfl_xor(idx,   msk, 32);
            if (s2 < score || (s2 == score && i2 < idx)) { score = s2; idx = i2; }
        }
        if (l16 == 0) { minV[m][wave] = score; minI[m][wave] = idx; }
    }
    __syncthreads();

    if (tid < 16) {
        float best = minV[tid][0];
        int   bi   = minI[tid][0];
#pragma unroll
        for (int w = 1; w < 16; ++w) {
            float s = minV[tid][w]; int i = minI[tid][w];
            if (s < best || (s == best && i < bi)) { best = s; bi = i; }
        }
        idxOut[m0 + tid]  = bi;
        idxOutF[m0 + tid] = (float)bi;   // indices slot of d_out (float dtype)
    }
}

// Kernel 4: gather winning code, quantized = R * code, per-block loss partial.
__global__ void quant_loss_kernel(const float* __restrict__ x,
                                  const float* __restrict__ codes,
                                  const float* __restrict__ u,
                                  const float* __restrict__ cArr,
                                  const int*  __restrict__ idx,
                                  float* __restrict__ quantOut,
                                  float* __restrict__ partials) {
    __shared__ float waveS[8];
    const int wave = threadIdx.x >> 5;
    const int lane = threadIdx.x & 31;
    const int b = blockIdx.x * 8 + wave;

    const int ci = idx[b];
    float4 q  = reinterpret_cast<const float4*>(codes + ci * DIM)[lane];
    float4 uu = reinterpret_cast<const float4*>(u + b * DIM)[lane];
    const float c = cArr[b];

    float s_u = waveSum(uu.x*q.x + uu.y*q.y + uu.z*q.z + uu.w*q.w);
    float s_v = waveSum(q.x + q.y + q.z + q.w) * INV_SQRT_D;
    float t = 1.0f + c + EPS_F;
    // R q = q + u*(s_v + (c*s_v - s_u)/t) + v*(-s_u + (c*s_u - s_v)/t)
    float cu = s_v + (c * s_v - s_u) / t;
    float cv = (-s_u + (c * s_u - s_v) / t) * INV_SQRT_D;

    float4 qr;
    qr.x = q.x + uu.x * cu + cv;
    qr.y = q.y + uu.y * cu + cv;
    qr.z = q.z + uu.z * cu + cv;
    qr.w = q.w + uu.w * cu + cv;
    reinterpret_cast<float4*>(quantOut + b * DIM)[lane] = qr;

    float4 xi = reinterpret_cast<const float4*>(x + b * DIM)[lane];
    float dx = xi.x - qr.x, dy = xi.y - qr.y, dz = xi.z - qr.z, dw = xi.w - qr.w;
    float d = waveSum(dx*dx + dy*dy + dz*dz + dw*dw);
    if (lane == 0) waveS[wave] = d;
    __syncthreads();
    if (threadIdx.x == 0) {
        float s = 0.0f;
#pragma unroll
        for (int w = 0; w < 8; ++w) s += waveS[w];
        partials[blockIdx.x] = s;
    }
}

// Kernel 5: fold 512 partials -> loss = (1+BETA) * mean_b ||x_b - q_b||^2
__global__ void loss_reduce_kernel(const float* __restrict__ partials,
                                   float* __restrict__ lossOut) {
    __shared__ float ws[16];
    float v = partials[threadIdx.x];
    v = waveSum(v);
    if ((threadIdx.x & 31) == 0) ws[threadIdx.x >> 5] = v;
    __syncthreads();
    if (threadIdx.x == 0) {
        float s = 0.0f;
#pragma unroll
        for (int w = 0; w < 16; ++w) s += ws[w];
        lossOut[0] = s * (1.25f / (float)B_ROWS);
    }
}

extern "C" void kernel_launch(void* const* d_in, const int* in_sizes, int n_in,
                              void* d_out, int out_size, void* d_ws, size_t ws_size,
                              hipStream_t stream) {
    const float* x     = (const float*)d_in[0];   // (4096,128)
    const float* pq    = (const float*)d_in[1];   // (4096,128)
    const float* codes = (const float*)d_in[2];   // (1,256,128)

    float* out = (float*)d_out;
    float* quantOut = out;                         // 524288 f32
    float* idxOutF  = out + B_ROWS * DIM;          // 4096 f32 (indices as float)
    float* lossOut  = out + B_ROWS * DIM + B_ROWS; // 1 f32

    // workspace layout (floats)
    float* ws   = (float*)d_ws;
    float* xc   = ws;                              // 524288
    float* u    = xc + B_ROWS * DIM;               // 524288
    float* cArr = u + B_ROWS * DIM;                // 4096
    int*   idx  = (int*)(cArr + B_ROWS);           // 4096 (int32)
    float* cn   = (float*)(idx + B_ROWS);          // 256
    float* part = cn + N_CODES;                    // 512

    // 1) canonicalize: xc = R^T x  (8 rows / 256-thread block)
    rot_canonical_kernel<<<B_ROWS / 8, 256, 0, stream>>>(x, pq, xc, u, cArr);
    // 2) code norms (one wave per code)
    code_norm_kernel<<<N_CODES, 32, 0, stream>>>(codes, cn);
    // 3) fp32 WMMA distance GEMM + argmin (16 rows / 512-thread block)
    dist_argmin_kernel<<<B_ROWS / 16, 512, 0, stream>>>(xc, codes, cn, idx, idxOutF);
    // 4) gather + rotate back + loss partials
    quant_loss_kernel<<<B_ROWS / 8, 256, 0, stream>>>(x, codes, u, cArr, idx,
                                                      quantOut, part);
    // 5) final loss
    loss_reduce_kernel<<<1, 512, 0, stream>>>(part, lossOut);
}